// SymmLoss_32581621908063
// MI455X (gfx1250) — compile-verified
//
#include <hip/hip_runtime.h>

typedef float v2f __attribute__((ext_vector_type(2)));
typedef float v8f __attribute__((ext_vector_type(8)));
typedef int   av4i __attribute__((vector_size(4 * sizeof(int))));   // matches async-LDS builtin proto
typedef __attribute__((address_space(1))) av4i* gv4i_ptr;           // global (AS1)
typedef __attribute__((address_space(3))) av4i* lv4i_ptr;           // LDS (AS3)

#define B_  4
#define N_  2048
#define M_  16
#define S_  200
#define SP_ 208   // S padded to 13*16

// workspace layout (float offsets)
#define OFF_PTSA 0                      // B*M*SP*4 = 53248  (A-ready float4: -2px,-2py,-2pz,|p|^2)
#define OFF_QT   53248                  // B*M*N*4  = 524288 (float4: qx,qy,qz,|q|^2)
#define OFF_D1   577536                 // B*N*M    = 131072
#define OFF_D2   708608                 // B*M*SP   = 13312
#define OFF_R    721920                 // B*M*9    = 576
#define OFF_AREA 722496                 // B*M      = 64
#define OFF_IDX  722560                 // B*M ints = 64
#define OFF_PART 722624                 // 32 partial sums (pcl_to_prim)
#define OFF_EMB  722656                 // 1 scalar (emb_reg)

#define HAS_ASYNC_LDS (__has_builtin(__builtin_amdgcn_global_load_async_to_lds_b128) && \
                       __has_builtin(__builtin_amdgcn_s_wait_asynccnt))

// ---------------------------------------------------------------------------
// Embeddings: idx_e = argmax over axis 1, plus emb_reg scalar. Deterministic.
// ---------------------------------------------------------------------------
__global__ void k_emb(const float* __restrict__ emb, float* __restrict__ W,
                      int* __restrict__ idx_e) {
  __shared__ float s1[64], s2[64], s3[256];
  int t = threadIdx.x;
  if (t < 64) {
    int b = t >> 4, j = t & 15;
    float cs = 0.f, best = -1e30f; int bi = 0;
    for (int i = 0; i < M_; ++i) {
      float v = emb[(b * M_ + i) * M_ + j];
      cs += v;
      if (v > best) { best = v; bi = i; }
    }
    idx_e[t] = bi;
    s1[t] = (cs - 1.f) * (cs - 1.f);
    int i = t & 15;
    float rs = 0.f;
    for (int j2 = 0; j2 < M_; ++j2) rs += emb[(b * M_ + i) * M_ + j2];
    s2[t] = (rs - 1.f) * (rs - 1.f);
  }
  float acc = 0.f;
  for (int e = t; e < B_ * M_ * M_; e += 256) { float v = emb[e]; acc += v * (1.f - v); }
  s3[t] = acc;
  __syncthreads();
  if (t == 0) {
    float t1 = 0.f, t2 = 0.f, t3 = 0.f;
    for (int k = 0; k < 64; ++k) { t1 += s1[k]; t2 += s2[k]; }
    for (int k = 0; k < 256; ++k) t3 += s3[k];
    W[OFF_EMB] = t1 / 64.f + 10.f * (t2 / 64.f) + t3 / (float)(B_ * M_ * M_);
  }
}

// ---------------------------------------------------------------------------
// Per (b, m_out): superquadric samples (gathered by idx_e), deform, pack
// WMMA-A-ready rows; also rotation matrices and raw areas.
// ---------------------------------------------------------------------------
__device__ __forceinline__ float fexp_(float v, float p) {
  float sg = (v > 0.f) ? 1.f : ((v < 0.f) ? -1.f : 0.f);
  return sg * powf(fabsf(v) + 1e-6f, p);
}

__global__ void k_prep(const float* __restrict__ size3,
                       const float* __restrict__ shape,
                       const float* __restrict__ deform,
                       const float* __restrict__ rot,
                       float* __restrict__ W, const int* __restrict__ idx_e) {
  int bm = blockIdx.x;            // b*M + m_out
  int b = bm >> 4;
  int ms = idx_e[bm];             // gathered source primitive
  int t = threadIdx.x;
  float4* ptsA = (float4*)(W + OFF_PTSA) + (size_t)bm * SP_;
  if (t < SP_) {
    if (t < S_) {
      const float PI = 3.14159265358979323846f;
      float eta = (-PI * 0.5f + 0.05f) + t * ((PI - 0.1f) / (float)(S_ - 1));
      float om  = (-PI + 0.05f)        + t * ((2.f * PI - 0.1f) / (float)(S_ - 1));
      float e1 = shape[(b * M_ + ms) * 2 + 0], e2 = shape[(b * M_ + ms) * 2 + 1];
      float sx = size3[(b * M_ + ms) * 3 + 0];
      float sy = size3[(b * M_ + ms) * 3 + 1];
      float sz = size3[(b * M_ + ms) * 3 + 2];
      float fce = fexp_(cosf(eta), e1);
      float x = sx * fce * fexp_(cosf(om), e2);
      float y = sy * fce * fexp_(sinf(om), e2);
      float z = sz * fexp_(sinf(eta), e1);
      float dfx = deform[(b * M_ + ms) * 2 + 0], dfy = deform[(b * M_ + ms) * 2 + 1];
      float px = x * (dfx / sz * z + 1.f);
      float py = y * (dfy / sz * z + 1.f);
      float pn = px * px + py * py + z * z;
      ptsA[t] = make_float4(-2.f * px, -2.f * py, -2.f * z, pn);
    } else {                      // padded rows: huge norm -> never the min over S
      ptsA[t] = make_float4(0.f, 0.f, 0.f, 1e30f);
    }
  }
  if (t == 0) {                   // rotation matrix (original m indexing)
    float qw = rot[bm * 4 + 0], qx = rot[bm * 4 + 1];
    float qy = rot[bm * 4 + 2], qz = rot[bm * 4 + 3];
    float inv = rsqrtf(qw * qw + qx * qx + qy * qy + qz * qz);
    qw *= inv; qx *= inv; qy *= inv; qz *= inv;
    float* R = W + OFF_R + bm * 9;
    R[0] = 1.f - 2.f * (qy * qy + qz * qz); R[1] = 2.f * (qx * qy - qw * qz); R[2] = 2.f * (qx * qz + qw * qy);
    R[3] = 2.f * (qx * qy + qw * qz); R[4] = 1.f - 2.f * (qx * qx + qz * qz); R[5] = 2.f * (qy * qz - qw * qx);
    R[6] = 2.f * (qx * qz - qw * qy); R[7] = 2.f * (qy * qz + qw * qx); R[8] = 1.f - 2.f * (qx * qx + qy * qy);
  }
  if (t == 32) {                  // raw area from gathered size
    float a0 = size3[(b * M_ + ms) * 3 + 0];
    float a1 = size3[(b * M_ + ms) * 3 + 1];
    float a2 = size3[(b * M_ + ms) * 3 + 2];
    float s = powf(a0 * a1, 1.6f) / 3.f + powf(a0 * a2, 1.6f) / 3.f + powf(a1 * a2, 1.6f) / 3.f;
    W[OFF_AREA + bm] = 4.f * 3.14159265358979323846f * powf(s, 0.625f);
  }
}

// ---------------------------------------------------------------------------
// pcl_t = R (p - t); store per (b,m,n) as float4: qx,qy,qz,|q|^2
// ---------------------------------------------------------------------------
__global__ void k_transform(const float* __restrict__ pcl,
                            const float* __restrict__ trans,
                            float* __restrict__ W) {
  int id = blockIdx.x * blockDim.x + threadIdx.x;   // b*N + n
  int b = id / N_, n = id % N_;
  float px = pcl[id * 3 + 0], py = pcl[id * 3 + 1], pz = pcl[id * 3 + 2];
  const float* Rb = W + OFF_R + b * M_ * 9;
  float4* qt = (float4*)(W + OFF_QT);
  for (int m = 0; m < M_; ++m) {
    float dx = px - trans[(b * M_ + m) * 3 + 0];
    float dy = py - trans[(b * M_ + m) * 3 + 1];
    float dz = pz - trans[(b * M_ + m) * 3 + 2];
    const float* R = Rb + m * 9;
    float qx = R[0] * dx + R[1] * dy + R[2] * dz;
    float qy = R[3] * dx + R[4] * dy + R[5] * dz;
    float qz = R[6] * dx + R[7] * dy + R[8] * dz;
    qt[(size_t)(b * M_ + m) * N_ + n] =
        make_float4(qx, qy, qz, qx * qx + qy * qy + qz * qz);
  }
}

// ---------------------------------------------------------------------------
// Core: per (b,m) workgroup, 8 waves. dist tile = WMMA(A=[-2p,|p|^2],
// B=[q,1], C=|q|^2). S-tile outer / column-tile inner:
//   d2 (min over N): per-lane v8f running min, flushed via ds_min_u32
//   d1 (min over S): per-tile v_min3 tree + all-lane ds_min_u32 per column
// No shuffles, no EXEC games; LDS uint-min is exact and order-independent.
// A rows staged to LDS via CDNA5 async load-to-LDS when available.
// ---------------------------------------------------------------------------
__global__ void __launch_bounds__(256) k_dist(float* __restrict__ W) {
  __shared__ float4 smA[SP_];        // 3.3 KB: A rows
  __shared__ unsigned smD1[N_];      // 8 KB: per-column min (d1)
  __shared__ unsigned smD2[SP_];     // 832 B: per-row min (d2)
  int bm = blockIdx.x; int b = bm >> 4, m = bm & 15;
  int t = threadIdx.x, wave = t >> 5, lane = t & 31;
  int lc = lane & 15;
  bool hi = lane >= 16;
  for (int i = t; i < N_; i += 256) smD1[i] = 0x7F800000u;   // +inf
  if (t < SP_) smD2[t] = 0x7F800000u;
  const float4* ptsA = (const float4*)(W + OFF_PTSA) + (size_t)bm * SP_;
#if HAS_ASYNC_LDS
  if (t < SP_) {
    __builtin_amdgcn_global_load_async_to_lds_b128(
        (gv4i_ptr)(ptsA + t), (lv4i_ptr)(&smA[t]), 0, 0);
  }
  __builtin_amdgcn_s_wait_asynccnt(0);
#else
  if (t < SP_) smA[t] = ptsA[t];
#endif
  __syncthreads();
  const float4* qt = (const float4*)(W + OFF_QT) + (size_t)bm * N_;
  for (int st = 0; st < 13; ++st) {
    float4 a = smA[st * 16 + lc];                      // ds_load_b128, hoisted
    v2f av = { hi ? a.z : a.x, hi ? a.w : a.y };
    v8f rowAcc = { 1e38f, 1e38f, 1e38f, 1e38f, 1e38f, 1e38f, 1e38f, 1e38f };
    for (int ct = wave; ct < N_ / 16; ct += 8) {
      int n = ct * 16 + lc;
      float4 q = qt[n];                                // global_load_b128
      v2f bv = { hi ? q.z : q.x, hi ? 1.0f : q.y };
      v8f c  = { q.w, q.w, q.w, q.w, q.w, q.w, q.w, q.w };
      v8f d = __builtin_amdgcn_wmma_f32_16x16x4_f32(
          false, av, false, bv, (short)0, c, false, false);
#pragma unroll
      for (int r = 0; r < 8; ++r) rowAcc[r] = fminf(rowAcc[r], d[r]);
      float lm = fminf(fminf(fminf(d[0], d[1]), fminf(d[2], d[3])),
                       fminf(fminf(d[4], d[5]), fminf(d[6], d[7])));
      atomicMin(&smD1[n], __float_as_uint(fmaxf(lm, 0.f)));   // all 32 lanes
    }
    int rbase = st * 16 + (hi ? 8 : 0);
#pragma unroll
    for (int r = 0; r < 8; ++r)
      atomicMin(&smD2[rbase + r], __float_as_uint(fmaxf(rowAcc[r], 0.f)));
  }
  __syncthreads();
  float* d1 = W + OFF_D1;
  for (int i = t; i < N_; i += 256)
    d1[((size_t)(b * N_) + i) * M_ + m] = __uint_as_float(smD1[i]);
  if (t < S_) W[OFF_D2 + (size_t)bm * SP_ + t] = __uint_as_float(smD2[t]);
}

// ---------------------------------------------------------------------------
// Per point: sort 16 (dist,prob) pairs, cumprod weighting, block partials.
// ---------------------------------------------------------------------------
__global__ void k_p2p(const float* __restrict__ prob, float* __restrict__ W) {
  __shared__ float red[256];
  int id = blockIdx.x * 256 + threadIdx.x;   // b*N + n
  int b = id / N_;
  const float* d1 = W + OFF_D1 + (size_t)id * M_;
  float dv[M_], pv[M_];
  for (int m = 0; m < M_; ++m) { dv[m] = d1[m]; pv[m] = prob[b * M_ + m]; }
  for (int i = 1; i < M_; ++i) {             // stable insertion sort (ascending)
    float dk = dv[i], pk = pv[i]; int j = i - 1;
    while (j >= 0 && dv[j] > dk) { dv[j + 1] = dv[j]; pv[j + 1] = pv[j]; --j; }
    dv[j + 1] = dk; pv[j + 1] = pk;
  }
  float neg = 1.f, sum = 0.f;
  for (int m = 0; m < M_; ++m) { sum += dv[m] * pv[m] * neg; neg *= (1.f - pv[m]); }
  red[threadIdx.x] = sum;
  __syncthreads();
  for (int s = 128; s > 0; s >>= 1) {
    if (threadIdx.x < s) red[threadIdx.x] += red[threadIdx.x + s];
    __syncthreads();
  }
  if (threadIdx.x == 0) W[OFF_PART + blockIdx.x] = red[0];
}

// ---------------------------------------------------------------------------
// Final: area normalization, prim_to_pcl, pcl_to_prim, write 5 scalars.
// ---------------------------------------------------------------------------
__global__ void k_final(float* __restrict__ W, float* __restrict__ out) {
  __shared__ float asum[4];
  __shared__ float val[64];
  int t = threadIdx.x;
  if (t < 4) {
    float s = 0.f;
    for (int m = 0; m < M_; ++m) s += W[OFF_AREA + t * M_ + m];
    asum[t] = s;
  }
  __syncthreads();
  if (t < 64) {
    int b = t >> 4;
    float area = (float)M_ * W[OFF_AREA + t] / asum[b];
    float s = 0.f;
    const float* d2 = W + OFF_D2 + (size_t)t * SP_;
    for (int k = 0; k < S_; ++k) { float v = d2[k]; s += (v >= 1e30f) ? 0.f : v; }
    val[t] = (s / (float)S_) * area;
  }
  __syncthreads();
  if (t == 0) {
    float prim = 0.f;
    for (int k = 0; k < 64; ++k) prim += val[k];
    prim /= (float)(B_ * M_);
    float p2p = 0.f;
    for (int k = 0; k < 32; ++k) p2p += W[OFF_PART + k];
    p2p /= (float)(B_ * N_);
    float embr = W[OFF_EMB];
    out[0] = p2p + prim + 0.f + embr;  // total
    out[1] = p2p;                      // pcl_to_prim
    out[2] = prim;                     // prim_to_pcl
    out[3] = 0.f;                      // regs
    out[4] = embr;                     // emb_reg
  }
}

extern "C" void kernel_launch(void* const* d_in, const int* in_sizes, int n_in,
                              void* d_out, int out_size, void* d_ws, size_t ws_size,
                              hipStream_t stream) {
  (void)in_sizes; (void)n_in; (void)out_size; (void)ws_size;
  const float* pcl    = (const float*)d_in[0];
  const float* trans  = (const float*)d_in[1];
  const float* rot    = (const float*)d_in[2];
  const float* size3  = (const float*)d_in[3];
  const float* shape  = (const float*)d_in[4];
  const float* deform = (const float*)d_in[5];
  const float* prob   = (const float*)d_in[6];
  const float* emb    = (const float*)d_in[7];
  float* W = (float*)d_ws;
  int* idx_e = (int*)(W + OFF_IDX);
  float* out = (float*)d_out;

  k_emb      <<<1,             256, 0, stream>>>(emb, W, idx_e);
  k_prep     <<<B_ * M_,       256, 0, stream>>>(size3, shape, deform, rot, W, idx_e);
  k_transform<<<(B_*N_)/256,   256, 0, stream>>>(pcl, trans, W);
  k_dist     <<<B_ * M_,       256, 0, stream>>>(W);
  k_p2p      <<<(B_*N_)/256,   256, 0, stream>>>(prob, W);
  k_final    <<<1,             256, 0, stream>>>(W, out);
}